// GuidedAttentionL1Loss_77481210020089
// MI455X (gfx1250) — compile-verified
//
#include <hip/hip_runtime.h>
#include <hip/hip_bf16.h>

// GuidedAttentionL1Loss for MI455X (gfx1250, wave32).
// Memory-bound: ~138 MB min traffic -> ~6us at 23.3 TB/s. One block per ragged
// segment; segment data staged into LDS once via global_load_async_to_lds_b128
// (CDNA5 async path, ASYNCcnt), then two LDS passes. Final combine uses
// v_wmma_f32_16x16x4_f32 with a ones B-matrix as a 64-elem/instr f32 reducer.

typedef __attribute__((ext_vector_type(2))) float v2f;
typedef __attribute__((ext_vector_type(8))) float v8f;

#define MAX_L 3072          // LDS tile capacity per array (12 KB each)
#define NLL_BLOCKS 32
#define ABS_BLOCKS 256

// ---------------- block reduction (blockDim == 256, 8 waves of 32) ----------
__device__ __forceinline__ float block_reduce256(float v, float* red) {
  #pragma unroll
  for (int o = 16; o > 0; o >>= 1) v += __shfl_xor(v, o, 32);
  const int wave = threadIdx.x >> 5;
  const int lane = threadIdx.x & 31;
  __syncthreads();                    // protect red from previous use
  if (lane == 0) red[wave] = v;
  __syncthreads();
  float t = 0.f;
  #pragma unroll
  for (int i = 0; i < 8; i++) t += red[i];   // LDS broadcast reads, fixed order
  return t;
}

// ---------------- prefix sum of lengths -> starts ---------------------------
__global__ __launch_bounds__(1024) void scan_starts_k(
    const int* __restrict__ lengths, int* __restrict__ starts, int B) {
  __shared__ int sums[1024];
  const int tid = threadIdx.x;
  const int per = (B + 1023) >> 10;
  const int base = tid * per;
  int s = 0;
  for (int i = 0; i < per; i++) { int j = base + i; if (j < B) s += lengths[j]; }
  sums[tid] = s;
  __syncthreads();
  for (int off = 1; off < 1024; off <<= 1) {
    int v = sums[tid];
    int add = (tid >= off) ? sums[tid - off] : 0;
    __syncthreads();
    sums[tid] = v + add;
    __syncthreads();
  }
  int run = (tid == 0) ? 0 : sums[tid - 1];
  for (int i = 0; i < per; i++) {
    int j = base + i;
    if (j < B) { starts[j] = run; run += lengths[j]; }
  }
}

// ---------------- NLL partials ----------------------------------------------
__global__ __launch_bounds__(256) void nll_partial_k(
    const float* __restrict__ logits, const int* __restrict__ labels,
    int B, float* __restrict__ out) {
  __shared__ float red[8];
  float acc = 0.f;
  const int stride = gridDim.x * blockDim.x;
  for (int i = blockIdx.x * blockDim.x + threadIdx.x; i < B; i += stride) {
    float l0 = logits[2 * i], l1 = logits[2 * i + 1];
    float m = fmaxf(l0, l1);
    float lse = m + logf(expf(l0 - m) + expf(l1 - m));
    float ly = labels[i] ? l1 : l0;
    acc += lse - ly;                 // -logp[label]
  }
  acc = block_reduce256(acc, red);
  if (threadIdx.x == 0) out[blockIdx.x] = acc;
}

// ---------------- L1 partials (float4 stream) -------------------------------
__global__ __launch_bounds__(256) void abs_partial_k(
    const float* __restrict__ p, int P, float* __restrict__ out) {
  __shared__ float red[8];
  float acc = 0.f;
  const int stride = gridDim.x * blockDim.x;
  const int P4 = P >> 2;
  const float4* p4 = (const float4*)p;
  for (int i = blockIdx.x * blockDim.x + threadIdx.x; i < P4; i += stride) {
    float4 v = p4[i];
    acc += fabsf(v.x) + fabsf(v.y) + fabsf(v.z) + fabsf(v.w);
  }
  for (int i = (P4 << 2) + blockIdx.x * blockDim.x + threadIdx.x; i < P; i += stride)
    acc += fabsf(p[i]);
  acc = block_reduce256(acc, red);
  if (threadIdx.x == 0) out[blockIdx.x] = acc;
}

// ---------------- per-segment two-pass compute ------------------------------
__device__ __forceinline__ void seg_compute(
    const float* __restrict__ pw, const float* __restrict__ px,
    int L, int lab, float* red, float* __restrict__ out_per) {
  const int tid = threadIdx.x;
  float sw = 0.f, sxw = 0.f, sw2 = 0.f;
  for (int e = tid; e < L; e += 256) {
    float wv = pw[e], xv = px[e];
    sw += wv;
    sxw = fmaf(xv, wv, sxw);
    sw2 = fmaf(wv, wv, sw2);
  }
  sw  = block_reduce256(sw,  red);
  sxw = block_reduce256(sxw, red);
  sw2 = block_reduce256(sw2, red);
  const float mean = sxw / sw;
  const float Lf = (float)L;
  const float ideal_std = (lab == 1 ? 1.0f : 1000.0f) / Lf;
  const float inv_std = 1.0f / ideal_std;
  const float coef = 0.3989422804014327f * inv_std;   // 1/(std*sqrt(2pi))
  float rs = 0.f, wr = 0.f, r2 = 0.f;
  for (int e = tid; e < L; e += 256) {
    float z  = (px[e] - mean) * inv_std;
    float rh = coef * expf(-0.5f * z * z);
    rs += rh;
    wr = fmaf(pw[e], rh, wr);
    r2 = fmaf(rh, rh, r2);
  }
  rs = block_reduce256(rs, red);
  wr = block_reduce256(wr, red);
  r2 = block_reduce256(r2, red);
  if (tid == 0) {
    float S = rs + 1e-6f;
    float invS = 1.0f / S;
    // sum((w - r_hat/S)^2) = sum(w^2) - 2*sum(w*r_hat)/S + sum(r_hat^2)/S^2
    *out_per = (sw2 - 2.0f * wr * invS + r2 * invS * invS) / Lf;
  }
}

__global__ __launch_bounds__(256) void seg_k(
    const float* __restrict__ w, const float* __restrict__ xpos,
    const int* __restrict__ labels, const int* __restrict__ lengths,
    const int* __restrict__ starts, float* __restrict__ awp_part) {
  __shared__ __align__(16) float s_w[MAX_L];
  __shared__ __align__(16) float s_x[MAX_L];
  __shared__ float red[8];
  const int b = blockIdx.x;
  const int L = lengths[b];
  const int start = starts[b];
  const int lab = labels[b];
  const int tid = threadIdx.x;

  if (L <= MAX_L) {
    // Stage segment into LDS with CDNA5 async loads (one global touch/byte).
    const float* gw = w + start;
    const float* gx = xpos + start;
    const int L4 = L & ~3;
    for (int e = tid * 4; e < L4; e += 256 * 4) {
      unsigned dw = (unsigned)(unsigned long long)&s_w[e];       // LDS byte offset
      unsigned long long aw = (unsigned long long)(gw + e);
      asm volatile("global_load_async_to_lds_b128 %0, %1, off"
                   :: "v"(dw), "v"(aw) : "memory");
      unsigned dx = (unsigned)(unsigned long long)&s_x[e];
      unsigned long long ax = (unsigned long long)(gx + e);
      asm volatile("global_load_async_to_lds_b128 %0, %1, off"
                   :: "v"(dx), "v"(ax) : "memory");
    }
    for (int e = L4 + tid; e < L; e += 256) { s_w[e] = gw[e]; s_x[e] = gx[e]; }
    asm volatile("s_wait_asynccnt 0x0" ::: "memory");   // this wave's async done
    __syncthreads();                                    // all waves' data visible
    seg_compute(s_w, s_x, L, lab, red, &awp_part[b]);
  } else {
    // Oversized-segment fallback: read directly from global (L2-resident).
    seg_compute(w + start, xpos + start, L, lab, red, &awp_part[b]);
  }
}

// ---------------- final combine: WMMA ones-reduction + scalar partials ------
__global__ __launch_bounds__(256) void combine_k(
    const float* __restrict__ nllp, const float* __restrict__ absp,
    const float* __restrict__ awpp, int B, float* __restrict__ out) {
  __shared__ float red[8];
  const int tid = threadIdx.x;
  const int lane = tid & 31;

  float a = 0.f;
  for (int i = tid; i < ABS_BLOCKS; i += 256) a += absp[i];
  const float abs_sum = block_reduce256(a, red);

  float n = 0.f;
  for (int i = tid; i < NLL_BLOCKS; i += 256) n += nllp[i];
  const float nll_sum = block_reduce256(n, red);

  // WMMA reduction of per-segment values: D = A(16x4) * ones(4x16) + C.
  // Each v_wmma_f32_16x16x4_f32 folds 64 f32 values; every column of D equals
  // the row-sum, so sum(D over wave) == 16 * sum(A). Uniform across all waves,
  // EXEC all-ones (256 threads = 8 full wave32s).
  v2f ones = {1.0f, 1.0f};
  v8f acc = {0.f, 0.f, 0.f, 0.f, 0.f, 0.f, 0.f, 0.f};
  const int B64 = B & ~63;
  for (int i = 0; i < B64; i += 64) {
    v2f av = {awpp[i + lane], awpp[i + 32 + lane]};
    acc = __builtin_amdgcn_wmma_f32_16x16x4_f32(
        /*neg_a=*/false, av, /*neg_b=*/false, ones,
        /*c_mod=*/(short)0, acc, /*reuse_a=*/false, /*reuse_b=*/false);
  }
  float s = acc[0] + acc[1] + acc[2] + acc[3] + acc[4] + acc[5] + acc[6] + acc[7];
  s *= (1.0f / 16.0f);                        // undo 16x column replication (exact)
  for (int i = B64 + lane; i < B; i += 32) s += awpp[i];   // tail if B%64 != 0
  #pragma unroll
  for (int o = 16; o > 0; o >>= 1) s += __shfl_xor(s, o, 32);

  if (tid == 0) {
    const float Bf = (float)B;
    const float nll = nll_sum / Bf;
    const float penalty = 5e-5f * abs_sum;     // ALPHA/2 * sum|p|
    const float awp = 0.5f * (s / Bf);         // BETA/2 * mean(per_seg)
    out[0] = nll + penalty + awp;
    out[1] = nll;
  }
}

// ---------------- launch ----------------------------------------------------
extern "C" void kernel_launch(void* const* d_in, const int* in_sizes, int n_in,
                              void* d_out, int out_size, void* d_ws, size_t ws_size,
                              hipStream_t stream) {
  const float* logits  = (const float*)d_in[0];   // [B,2]
  const int*   labels  = (const int*)  d_in[1];   // [B]
  const float* attw    = (const float*)d_in[2];   // [T]
  const float* params  = (const float*)d_in[3];   // [P]
  const float* xpos    = (const float*)d_in[4];   // [T]
  // d_in[5] segment_ids unused: block-per-segment makes it redundant (67 MB saved)
  const int*   lengths = (const int*)  d_in[6];   // [B]

  const int B = in_sizes[1];
  const int P = in_sizes[3];

  // workspace layout (floats): [awp partials: B][nll: 32][abs: 256][starts(int): B]
  float* ws = (float*)d_ws;
  float* awp_part = ws;
  float* nll_part = ws + B;
  float* abs_part = ws + B + NLL_BLOCKS;
  int*   starts   = (int*)(ws + B + NLL_BLOCKS + ABS_BLOCKS);

  scan_starts_k<<<1, 1024, 0, stream>>>(lengths, starts, B);
  nll_partial_k<<<NLL_BLOCKS, 256, 0, stream>>>(logits, labels, B, nll_part);
  abs_partial_k<<<ABS_BLOCKS, 256, 0, stream>>>(params, P, abs_part);
  seg_k<<<B, 256, 0, stream>>>(attw, xpos, labels, lengths, starts, awp_part);
  combine_k<<<1, 256, 0, stream>>>(nll_part, abs_part, awp_part, B, (float*)d_out);
}